// GCN_7627861917707
// MI455X (gfx1250) — compile-verified
//
#include <hip/hip_runtime.h>
#include <hip/hip_bf16.h>

typedef __attribute__((ext_vector_type(2))) float v2f;
typedef __attribute__((ext_vector_type(8))) float v8f;

#define NUM_NODES 30000
#define NUM_EDGES 480000
#define NFEAT 1024
#define SCAN_NB ((NUM_NODES + 255) / 256)   // 118

// ---------------------------------------------------------------- CSR build

__global__ void zero_i_kernel(int* __restrict__ p, int n) {
    int i = blockIdx.x * blockDim.x + threadIdx.x;
    if (i < n) p[i] = 0;
}

__global__ void count_kernel(const int* __restrict__ edges, int* __restrict__ cnt, int ne) {
    int e = blockIdx.x * blockDim.x + threadIdx.x;
    if (e < ne) atomicAdd(&cnt[edges[2 * e]], 1);
}

__global__ void count_sums_kernel(const int* __restrict__ cnt, int* __restrict__ bsum, int n) {
    __shared__ int sh[256];
    int i = blockIdx.x * 256 + threadIdx.x;
    sh[threadIdx.x] = (i < n) ? cnt[i] : 0;
    __syncthreads();
#pragma unroll
    for (int s = 128; s > 0; s >>= 1) {
        if (threadIdx.x < s) sh[threadIdx.x] += sh[threadIdx.x + s];
        __syncthreads();
    }
    if (threadIdx.x == 0) bsum[blockIdx.x] = sh[0];
}

__global__ void scan_bsum_kernel(int* __restrict__ bsum, int nb) {
    if (blockIdx.x == 0 && threadIdx.x == 0) {
        int acc = 0;
        for (int i = 0; i < nb; ++i) {
            int t = bsum[i];
            bsum[i] = acc;
            acc += t;
        }
    }
}

__global__ void scan_final_kernel(const int* __restrict__ cnt, const int* __restrict__ bsum,
                                  int* __restrict__ row_start, float* __restrict__ inv, int n) {
    __shared__ int sh[256];
    int i = blockIdx.x * 256 + threadIdx.x;
    int v = (i < n) ? cnt[i] : 0;
    sh[threadIdx.x] = v;
    __syncthreads();
#pragma unroll
    for (int s = 1; s < 256; s <<= 1) {
        int t = (threadIdx.x >= (unsigned)s) ? sh[threadIdx.x - s] : 0;
        __syncthreads();
        sh[threadIdx.x] += t;
        __syncthreads();
    }
    if (i < n) {
        row_start[i] = bsum[blockIdx.x] + sh[threadIdx.x] - v;  // exclusive prefix
        inv[i] = (v > 0) ? (1.0f / (float)v) : 0.f;
    }
}

__global__ void fill_csr_kernel(const int* __restrict__ edges, const int* __restrict__ row_start,
                                int* __restrict__ cursor, int* __restrict__ csr_col, int ne) {
    int e = blockIdx.x * blockDim.x + threadIdx.x;
    if (e < ne) {
        int r = edges[2 * e];
        int c = edges[2 * e + 1];
        int pos = atomicAdd(&cursor[r], 1);
        csr_col[row_start[r] + pos] = c;
    }
}

// sort each row's column list -> deterministic summation order across replays
__global__ void sort_rows_kernel(int* __restrict__ csr_col, const int* __restrict__ row_start,
                                 const int* __restrict__ cnt, int n) {
    int i = blockIdx.x * blockDim.x + threadIdx.x;
    if (i >= n) return;
    int beg = row_start[i];
    int m = cnt[i];
    for (int a = 1; a < m; ++a) {
        int key = csr_col[beg + a];
        int b = a - 1;
        while (b >= 0 && csr_col[beg + b] > key) {
            csr_col[beg + b + 1] = csr_col[beg + b];
            --b;
        }
        csr_col[beg + b + 1] = key;
    }
}

// ---------------------------------------------------------------- SpMM (CSR gather)
// g[i] = h[i] + inv_deg[i] * sum_{e in row i} h[csr_col[e]]
// block per node, blockDim = K/4, 4-way edge unroll for memory-level parallelism.
__global__ void spmm_kernel(const float* __restrict__ h, const int* __restrict__ row_start,
                            const int* __restrict__ cnt, const int* __restrict__ csr_col,
                            const float* __restrict__ inv, float* __restrict__ g, int K) {
    int i = blockIdx.x;
    int f = threadIdx.x * 4;
    int beg = row_start[i];
    int end = beg + cnt[i];
    float4 s = make_float4(0.f, 0.f, 0.f, 0.f);
    int e = beg;
    for (; e + 4 <= end; e += 4) {
        int c0 = csr_col[e + 0], c1 = csr_col[e + 1];
        int c2 = csr_col[e + 2], c3 = csr_col[e + 3];
        float4 v0 = *(const float4*)(h + (size_t)c0 * K + f);
        float4 v1 = *(const float4*)(h + (size_t)c1 * K + f);
        float4 v2 = *(const float4*)(h + (size_t)c2 * K + f);
        float4 v3 = *(const float4*)(h + (size_t)c3 * K + f);
        s.x += v0.x + v1.x + v2.x + v3.x;
        s.y += v0.y + v1.y + v2.y + v3.y;
        s.z += v0.z + v1.z + v2.z + v3.z;
        s.w += v0.w + v1.w + v2.w + v3.w;
    }
    for (; e < end; ++e) {
        int c = csr_col[e];
        float4 v = *(const float4*)(h + (size_t)c * K + f);
        s.x += v.x; s.y += v.y; s.z += v.z; s.w += v.w;
    }
    float w = inv[i];
    float4 hv = *(const float4*)(h + (size_t)i * K + f);
    float4 o;
    o.x = hv.x + w * s.x;
    o.y = hv.y + w * s.y;
    o.z = hv.z + w * s.z;
    o.w = hv.w + w * s.w;
    *(float4*)(g + (size_t)i * K + f) = o;
}

// ---------------------------------------------------------------- WMMA GEMM
// C[M,N] = A[M,K] @ B[K,N] + bias, optional leaky relu (slope 0.2).
// Block: 256 threads (8 waves). Tile: 128(M) x 64(N), K staged by 32,
// register-double-buffered global->LDS staging. Exact f32 via V_WMMA_F32_16X16X4_F32.

#define TM 128
#define TN 64
#define KT 32
#define AS_STRIDE 36  // multiple of 4 (float4-aligned stores), conflict-free frag reads
#define BS_STRIDE 34  // even (float2-aligned reads), conflict-free frag reads

__global__ __launch_bounds__(256) void gemm_wmma_kernel(
    const float* __restrict__ A, const float* __restrict__ B,
    const float* __restrict__ bias, float* __restrict__ C,
    int M, int K, int N, int relu) {
    __shared__ __align__(16) float As[TM][AS_STRIDE];
    __shared__ __align__(16) float Bs[TN][BS_STRIDE];  // transposed: Bs[n][k]

    const int tid = threadIdx.x;
    const int wave = tid >> 5;
    const int lane = tid & 31;
    const int lane16 = lane & 15;
    const int hi2 = (lane >> 4) << 1;
    const int m0 = blockIdx.x * TM;
    const int n0 = blockIdx.y * TN;

    float4 ra[4], rb[2];

    auto load_tile = [&](int kt) {
#pragma unroll
        for (int i = 0; i < 4; ++i) {
            int flat = tid + i * 256;
            int r = flat >> 3;
            int c = (flat & 7) << 2;
            int gr = m0 + r;
            ra[i] = make_float4(0.f, 0.f, 0.f, 0.f);
            if (gr < M) ra[i] = *(const float4*)(A + (size_t)gr * K + kt + c);
        }
#pragma unroll
        for (int i = 0; i < 2; ++i) {
            int flat = tid + i * 256;
            int r = flat >> 4;
            int c = (flat & 15) << 2;
            rb[i] = *(const float4*)(B + (size_t)(kt + r) * N + n0 + c);
        }
    };
    auto store_tile = [&]() {
#pragma unroll
        for (int i = 0; i < 4; ++i) {
            int flat = tid + i * 256;
            int r = flat >> 3;
            int c = (flat & 7) << 2;
            *(float4*)(&As[r][c]) = ra[i];
        }
#pragma unroll
        for (int i = 0; i < 2; ++i) {
            int flat = tid + i * 256;
            int r = flat >> 4;
            int c = (flat & 15) << 2;
            Bs[c + 0][r] = rb[i].x;
            Bs[c + 1][r] = rb[i].y;
            Bs[c + 2][r] = rb[i].z;
            Bs[c + 3][r] = rb[i].w;
        }
    };

    v8f acc[4] = {};
    load_tile(0);
    store_tile();

    for (int kt = 0; kt < K; kt += KT) {
        __syncthreads();  // LDS tile ready for all waves
        const bool has_next = (kt + KT) < K;
        if (has_next) load_tile(kt + KT);  // issue next-tile global loads now

        const int mrow = 16 * wave + lane16;
#pragma unroll
        for (int k = 0; k < KT; k += 4) {
            v2f a = *(const v2f*)(&As[mrow][k + hi2]);
#pragma unroll
            for (int t = 0; t < 4; ++t) {
                v2f b = *(const v2f*)(&Bs[16 * t + lane16][k + hi2]);
                acc[t] = __builtin_amdgcn_wmma_f32_16x16x4_f32(
                    false, a, false, b, (short)0, acc[t], false, false);
            }
        }
        __syncthreads();  // all waves done reading this tile
        if (has_next) store_tile();
    }

    // epilogue: bias + leaky relu; C layout M = v + 8*(lane/16), N = lane%16
    const int hi = lane >> 4;
#pragma unroll
    for (int t = 0; t < 4; ++t) {
        int n = n0 + 16 * t + lane16;
        float bv = bias[n];
#pragma unroll
        for (int v = 0; v < 8; ++v) {
            int m = m0 + 16 * wave + v + 8 * hi;
            if (m < M) {
                float o = acc[t][v] + bv;
                if (relu) o = (o >= 0.f) ? o : 0.2f * o;
                C[(size_t)m * N + n] = o;
            }
        }
    }
}

// ---------------------------------------------------------------- L2 normalize rows

__global__ void normalize_kernel(const float* __restrict__ h, float* __restrict__ out, int K) {
    int row = blockIdx.x;
    int tid = threadIdx.x;  // 256 threads, K = 1024
    const float* hr = h + (size_t)row * K;
    float4 v = *(const float4*)(hr + tid * 4);
    float s = v.x * v.x + v.y * v.y + v.z * v.z + v.w * v.w;
#pragma unroll
    for (int off = 16; off > 0; off >>= 1) s += __shfl_xor(s, off, 32);
    __shared__ float warp_s[8];
    __shared__ float inv_norm;
    if ((tid & 31) == 0) warp_s[tid >> 5] = s;
    __syncthreads();
    if (tid == 0) {
        float t = 0.f;
#pragma unroll
        for (int i = 0; i < 8; ++i) t += warp_s[i];
        inv_norm = 1.0f / fmaxf(sqrtf(t), 1e-12f);
    }
    __syncthreads();
    float inv = inv_norm;
    v.x *= inv; v.y *= inv; v.z *= inv; v.w *= inv;
    *(float4*)(out + (size_t)row * K + tid * 4) = v;
}

// ---------------------------------------------------------------- launch

extern "C" void kernel_launch(void* const* d_in, const int* in_sizes, int n_in,
                              void* d_out, int out_size, void* d_ws, size_t ws_size,
                              hipStream_t stream) {
    const float* x  = (const float*)d_in[0];
    const float* w1 = (const float*)d_in[1];
    const float* b1 = (const float*)d_in[2];
    const float* w2 = (const float*)d_in[3];
    const float* b2 = (const float*)d_in[4];
    const float* w3 = (const float*)d_in[5];
    const float* b3 = (const float*)d_in[6];
    const int* edges = (const int*)d_in[7];
    float* out = (float*)d_out;

    const long long featBytes = (long long)NUM_NODES * NFEAT * sizeof(float);  // 122,880,000
    char* ws = (char*)d_ws;
    float* gbuf = (float*)(ws);
    float* hA   = (float*)(ws + featBytes);
    float* hB   = (float*)(ws + 2 * featBytes);
    char*  p    = ws + 3 * featBytes;
    int*   cnt       = (int*)(p);             p += 120000;
    int*   cursor    = (int*)(p);             p += 120000;
    int*   row_start = (int*)(p);             p += 120000;
    int*   csr_col   = (int*)(p);             p += 1920000;
    float* inv       = (float*)(p);           p += 120000;
    int*   bsum      = (int*)(p);

    // ---- build CSR (deterministic after per-row sort) + inverse degrees
    zero_i_kernel<<<(NUM_NODES + 255) / 256, 256, 0, stream>>>(cnt, NUM_NODES);
    zero_i_kernel<<<(NUM_NODES + 255) / 256, 256, 0, stream>>>(cursor, NUM_NODES);
    count_kernel<<<(NUM_EDGES + 255) / 256, 256, 0, stream>>>(edges, cnt, NUM_EDGES);
    count_sums_kernel<<<SCAN_NB, 256, 0, stream>>>(cnt, bsum, NUM_NODES);
    scan_bsum_kernel<<<1, 32, 0, stream>>>(bsum, SCAN_NB);
    scan_final_kernel<<<SCAN_NB, 256, 0, stream>>>(cnt, bsum, row_start, inv, NUM_NODES);
    fill_csr_kernel<<<(NUM_EDGES + 255) / 256, 256, 0, stream>>>(edges, row_start, cursor,
                                                                 csr_col, NUM_EDGES);
    sort_rows_kernel<<<(NUM_NODES + 255) / 256, 256, 0, stream>>>(csr_col, row_start, cnt,
                                                                  NUM_NODES);

    auto layer = [&](const float* hin, const float* w, const float* b, float* hout,
                     int K, int relu) {
        spmm_kernel<<<NUM_NODES, K / 4, 0, stream>>>(hin, row_start, cnt, csr_col, inv, gbuf, K);
        dim3 grid((NUM_NODES + TM - 1) / TM, NFEAT / TN);
        gemm_wmma_kernel<<<grid, 256, 0, stream>>>(gbuf, w, b, hout, NUM_NODES, K, NFEAT, relu);
    };

    layer(x,  w1, b1, hA, 512,   1);
    layer(hA, w2, b2, hB, NFEAT, 1);
    layer(hB, w3, b3, hA, NFEAT, 0);

    normalize_kernel<<<NUM_NODES, 256, 0, stream>>>(hA, out, NFEAT);
}